// ODEINDLayer_2843268350253
// MI455X (gfx1250) — compile-verified
//
#include <hip/hip_runtime.h>
#include <hip/hip_bf16.h>

typedef float v2f __attribute__((ext_vector_type(2)));
typedef float v8f __attribute__((ext_vector_type(8)));

#define SA    104   // LDS row stride in floats (bank-conflict-free for k/k+2 pairs)
#define KROWS 180   // 178 rows of A, padded to multiple of 4
#define NB    90    // normal-equation dimension
#define BW    6     // half bandwidth of AtA (and of its Cholesky factor)

__global__ __launch_bounds__(256)
void odein_solve_kernel(const float* __restrict__ coeffs,  // (BS,30,1,1,3)
                        const float* __restrict__ rhs,     // (BS,30,1)
                        const float* __restrict__ iv_rhs,  // (BS,2,1,2)
                        const float* __restrict__ steps,   // (BS,29,1)
                        float* __restrict__ out)           // (BS,30,1,3)
{
    __shared__ float Abuf[KROWS * SA];   // A (178x90 padded); reused as AtA (96 x SA)
    __shared__ float st[32];
    __shared__ float r1[96];             // Aᵀβ, then y, then x

    const int tid = threadIdx.x;
    const int b   = blockIdx.x;

    // ---- zero A, stage steps -------------------------------------------------
    for (int idx = tid; idx < KROWS * SA; idx += 256) Abuf[idx] = 0.0f;
    if (tid < 29) st[tid] = steps[b * 29 + tid];
    if (tid < 96) r1[tid] = 0.0f;
    __syncthreads();

    // ---- fill sparse A + rhs1 = Aᵀβ (first 34 rows only) ---------------------
    if (tid < 90) {                       // collocation rows 0..29
        float c = coeffs[b * 90 + tid];
        int s = tid / 3;
        Abuf[s * SA + tid] = c;
        float v = c * rhs[b * 30 + s];
        if      (tid == 0) v += iv_rhs[b * 4 + 0];
        else if (tid == 1) v += iv_rhs[b * 4 + 1];
        else if (tid == 3) v += iv_rhs[b * 4 + 2];
        else if (tid == 4) v += iv_rhs[b * 4 + 3];
        r1[tid] = v;
    }
    if (tid >= 96 && tid < 100) {         // iv rows 30..33: ones at cols 0,1,3,4
        const int ivcol[4] = {0, 1, 3, 4};
        Abuf[(30 + (tid - 96)) * SA + ivcol[tid - 96]] = 1.0f;
    }
    if (tid >= 128 && tid < 157) {        // fwd rows 34..91, bwd rows 120..177
        int s = tid - 128;                // step 0..28
        float h = st[s], h2 = h * h;
        int r = 34 + 2 * s;               // fwd i=0
        Abuf[r * SA + 3 * (s + 1) + 0] = -1.0f;
        Abuf[r * SA + 3 * s + 0] = 1.0f;
        Abuf[r * SA + 3 * s + 1] = h;
        Abuf[r * SA + 3 * s + 2] = 0.5f * h2;
        r = 35 + 2 * s;                   // fwd i=1
        Abuf[r * SA + 3 * (s + 1) + 1] = -h;
        Abuf[r * SA + 3 * s + 1] = h;
        Abuf[r * SA + 3 * s + 2] = h2;
        r = 120 + 2 * s;                  // bwd i=0
        Abuf[r * SA + 3 * s + 0] = -1.0f;
        Abuf[r * SA + 3 * (s + 1) + 0] = 1.0f;
        Abuf[r * SA + 3 * (s + 1) + 1] = -h;
        Abuf[r * SA + 3 * (s + 1) + 2] = 0.5f * h2;
        r = 121 + 2 * s;                  // bwd i=1
        Abuf[r * SA + 3 * s + 1] = h;
        Abuf[r * SA + 3 * (s + 1) + 1] = -h;
        Abuf[r * SA + 3 * (s + 1) + 2] = h2;
    }
    if (tid >= 160 && tid < 188) {        // central rows 92..119 (step 1..28)
        int s = tid - 159;
        int r = 91 + s;
        Abuf[r * SA + 3 * (s - 1) + 1] = -1.0f;
        Abuf[r * SA + 3 * (s + 1) + 1] = 1.0f;
        Abuf[r * SA + 3 * s + 2] = -(st[s] + st[s - 1]);
    }
    __syncthreads();

    // ---- AtA = AᵀA on the 16 block-tridiagonal tiles via V_WMMA_F32_16X16X4_F32
    const int lane = tid & 31, wv = tid >> 5;
    const int half = lane >> 4, lm = lane & 15;
    const int TI[16] = {0,0, 1,1,1, 2,2,2, 3,3,3, 4,4,4, 5,5};
    const int TJ[16] = {0,1, 0,1,2, 1,2,3, 2,3,4, 3,4,5, 4,5};
    const int m0a = TI[2 * wv] * 16,     n0a = TJ[2 * wv] * 16;
    const int m0b = TI[2 * wv + 1] * 16, n0b = TJ[2 * wv + 1] * 16;

    v8f acc0 = {0.f, 0.f, 0.f, 0.f, 0.f, 0.f, 0.f, 0.f};
    v8f acc1 = {0.f, 0.f, 0.f, 0.f, 0.f, 0.f, 0.f, 0.f};
    for (int kk = 0; kk < KROWS / 4; ++kk) {
        int k0 = kk * 4 + 2 * half;       // lanes 0-15: K=0,1  lanes 16-31: K=2,3
        v2f a, bb;
        a.x  = Abuf[k0 * SA + m0a + lm];
        a.y  = Abuf[(k0 + 1) * SA + m0a + lm];
        bb.x = Abuf[k0 * SA + n0a + lm];
        bb.y = Abuf[(k0 + 1) * SA + n0a + lm];
        acc0 = __builtin_amdgcn_wmma_f32_16x16x4_f32(false, a, false, bb,
                                                     (short)0, acc0, false, false);
        a.x  = Abuf[k0 * SA + m0b + lm];
        a.y  = Abuf[(k0 + 1) * SA + m0b + lm];
        bb.x = Abuf[k0 * SA + n0b + lm];
        bb.y = Abuf[(k0 + 1) * SA + n0b + lm];
        acc1 = __builtin_amdgcn_wmma_f32_16x16x4_f32(false, a, false, bb,
                                                     (short)0, acc1, false, false);
    }
    __syncthreads();                      // all reads of A done; reuse buffer for AtA
    float* AtA = Abuf;
#pragma unroll
    for (int v = 0; v < 8; ++v)           // C layout: M = v + 8*half, N = lm
        AtA[(m0a + v + 8 * half) * SA + n0a + lm] = acc0[v];
#pragma unroll
    for (int v = 0; v < 8; ++v)
        AtA[(m0b + v + 8 * half) * SA + n0b + lm] = acc1[v];
    __syncthreads();

    // ---- band Cholesky (half-bandwidth 6), in place --------------------------
    const int DR[21] = {1,2,3,4,5,6, 2,3,4,5,6, 3,4,5,6, 4,5,6, 5,6, 6};
    const int DC[21] = {1,1,1,1,1,1, 2,2,2,2,2, 3,3,3,3, 4,4,4, 5,5, 6};
    for (int j = 0; j < NB; ++j) {
        if (tid == 0) {
            float d = sqrtf(AtA[j * SA + j]);
            AtA[j * SA + j] = d;
            float inv = 1.0f / d;
            int rmax = (j + BW < NB - 1) ? (j + BW) : (NB - 1);
            for (int r = j + 1; r <= rmax; ++r) AtA[r * SA + j] *= inv;
        }
        __syncthreads();
        if (tid < 21) {
            int r = j + DR[tid];
            if (r < NB) {
                int c = j + DC[tid];
                AtA[r * SA + c] -= AtA[r * SA + j] * AtA[c * SA + j];
            }
        }
        __syncthreads();
    }

    // ---- band forward/back substitution --------------------------------------
    if (tid == 0) {
        for (int j = 0; j < NB; ++j) {    // L y = rhs1
            float s = r1[j];
            int k0 = (j - BW > 0) ? (j - BW) : 0;
            for (int k = k0; k < j; ++k) s -= AtA[j * SA + k] * r1[k];
            r1[j] = s / AtA[j * SA + j];
        }
        for (int j = NB - 1; j >= 0; --j) {  // Lᵀ x = y
            float s = r1[j];
            int k1 = (j + BW < NB - 1) ? (j + BW) : (NB - 1);
            for (int k = j + 1; k <= k1; ++k) s -= AtA[k * SA + j] * r1[k];
            r1[j] = s / AtA[j * SA + j];
        }
    }
    __syncthreads();
    if (tid < 90) out[b * 90 + tid] = r1[tid];
}

extern "C" void kernel_launch(void* const* d_in, const int* in_sizes, int n_in,
                              void* d_out, int out_size, void* d_ws, size_t ws_size,
                              hipStream_t stream) {
    (void)n_in; (void)out_size; (void)d_ws; (void)ws_size;
    const float* coeffs = (const float*)d_in[0];
    const float* rhs    = (const float*)d_in[1];
    const float* iv     = (const float*)d_in[2];
    const float* steps  = (const float*)d_in[3];
    float* out = (float*)d_out;
    int bs = in_sizes[0] / 90;            // 8192
    odein_solve_kernel<<<bs, 256, 0, stream>>>(coeffs, rhs, iv, steps, out);
}